// LightGCN_8942121910853
// MI455X (gfx1250) — compile-verified
//
#include <hip/hip_runtime.h>

// LightGCN on MI455X (gfx1250, wave32).
//  - SpMM layers: L2-resident (38.4MB per emb buffer, 192MB L2), float4 gathers +
//    native f32 atomic scatter (unsafeAtomicAdd -> global_atomic_add_f32).
//  - Final batched dot: V_WMMA_F32_16X16X4_F32, 16 chained WMMAs per 16-batch tile,
//    diagonal extraction per the ISA 16x16 f32 C/D VGPR layout (no scratch spill:
//    compile-time-indexed cndmask select chain instead of dynamic vector subscript).

typedef __attribute__((ext_vector_type(2))) float v2f;
typedef __attribute__((ext_vector_type(8))) float v8f;

#define EMB_D 64

// ---------------------------------------------------------------------------
// Kernel 1: cur = acc = concat(user_emb, item_emb), vectorized float4
// ---------------------------------------------------------------------------
__global__ void lgcn_init(const float4* __restrict__ ue4,
                          const float4* __restrict__ ie4,
                          float4* __restrict__ cur,
                          float4* __restrict__ acc,
                          long long nU4, long long nTot4) {
    long long t = (long long)blockIdx.x * blockDim.x + threadIdx.x;
    if (t >= nTot4) return;
    float4 v = (t < nU4) ? ue4[t] : ie4[t - nU4];
    cur[t] = v;
    acc[t] = v;
}

// ---------------------------------------------------------------------------
// Kernel 2: COO scatter  dst[row] += val * src[col]
// 16 lanes per edge; each lane owns one float4 chunk of the 64-wide row.
// Gather = global_load_b128 (coalesced); scatter = global_atomic_add_f32.
// ---------------------------------------------------------------------------
__global__ void lgcn_spmm_scatter(const int* __restrict__ rows,
                                  const int* __restrict__ cols,
                                  const float* __restrict__ vals,
                                  const float* __restrict__ src,
                                  float* __restrict__ dst,
                                  long long nnz) {
    long long t = (long long)blockIdx.x * blockDim.x + threadIdx.x;
    long long e = t >> 4;
    if (e >= nnz) return;
    int c = (int)(t & 15);

    int   row = rows[e];
    int   col = cols[e];
    float v   = vals[e];

    float4 s = ((const float4*)(src + (long long)col * EMB_D))[c];
    float* d = dst + (long long)row * EMB_D + c * 4;

    unsafeAtomicAdd(d + 0, v * s.x);
    unsafeAtomicAdd(d + 1, v * s.y);
    unsafeAtomicAdd(d + 2, v * s.z);
    unsafeAtomicAdd(d + 3, v * s.w);
}

// ---------------------------------------------------------------------------
// Kernel 3: acc += next (float4)
// ---------------------------------------------------------------------------
__global__ void lgcn_acc_add(float4* __restrict__ acc,
                             const float4* __restrict__ nxt,
                             long long n4) {
    long long t = (long long)blockIdx.x * blockDim.x + threadIdx.x;
    if (t >= n4) return;
    float4 a = acc[t];
    float4 b = nxt[t];
    a.x += b.x; a.y += b.y; a.z += b.z; a.w += b.w;
    acc[t] = a;
}

// ---------------------------------------------------------------------------
// Kernel 4: gamma[b] = dot(acc[users[b]], acc[n_users+items[b]]) / 16
// (the 1/16 folds both 1/(LAYERS+1) factors of the two operands)
//
// One wave32 per 16-batch tile. Computes the full 16x16 tile U*I^T with
// 16 chained V_WMMA_F32_16X16X4_F32 ops (K=64), extracts the diagonal.
//
// Operand layout (ISA 7.12.2, 32-bit A 16x4): lane L<16 holds M=L, K={k0,k0+1};
// lanes 16-31 hold M=L-16, K={k0+2,k0+3}. B (4x16) is symmetric.
// C/D 16x16 f32: element (M,N): M<8 -> VGPR M, lane N; M>=8 -> VGPR M-8, lane N+16.
// Diagonal: lanes 0-7 -> c[lane]; lanes 24-31 -> c[lane-24].
// ---------------------------------------------------------------------------
__global__ void lgcn_dot_wmma(const int* __restrict__ users,
                              const int* __restrict__ items,
                              const float* __restrict__ acc,
                              float* __restrict__ out,
                              int n_users, int batch) {
    int tile = blockIdx.x;
    int lane = threadIdx.x;          // 0..31, full wave active (EXEC all ones)
    int r    = lane & 15;            // M (for A) / N (for B)
    int koff = (lane >> 4) * 2;      // K sub-slice owned by this half-wave

    int b = tile * 16 + r;
    if (b >= batch) b = batch - 1;   // clamp reads; writes are guarded below

    const float* up = acc + (long long)users[b] * EMB_D;
    const float* ip = acc + ((long long)items[b] + n_users) * EMB_D;

    v8f c = {0.f, 0.f, 0.f, 0.f, 0.f, 0.f, 0.f, 0.f};
#pragma unroll
    for (int k0 = 0; k0 < EMB_D; k0 += 4) {
        v2f a, bm;
        a.x  = up[k0 + koff];
        a.y  = up[k0 + koff + 1];
        bm.x = ip[k0 + koff];
        bm.y = ip[k0 + koff + 1];
        c = __builtin_amdgcn_wmma_f32_16x16x4_f32(
                /*neg_a=*/false, a, /*neg_b=*/false, bm,
                /*c_mod=*/(short)0, c, /*reuse_a=*/false, /*reuse_b=*/false);
    }

    // Diagonal element index within this lane's VGPR stack (if any):
    //   lanes 0-7  -> component lane   (M = lane)
    //   lanes 24-31-> component lane-24 (M = lane-16)
    int  idx  = (lane < 8) ? lane : (lane - 24);
    bool have = (lane < 8) | (lane >= 24);
    int  m    = (lane < 8) ? lane : (lane - 16);

    // Compile-time-indexed select chain: no scratch, just v_cndmask ops.
    float diag = 0.f;
#pragma unroll
    for (int j = 0; j < 8; ++j) {
        diag = (idx == j) ? c[j] : diag;
    }

    int ob = tile * 16 + m;
    if (have && ob < batch) out[ob] = diag * (1.0f / 16.0f);
}

// ---------------------------------------------------------------------------
extern "C" void kernel_launch(void* const* d_in, const int* in_sizes, int n_in,
                              void* d_out, int out_size, void* d_ws, size_t ws_size,
                              hipStream_t stream) {
    const int*   users    = (const int*)d_in[0];
    const int*   items    = (const int*)d_in[1];
    const float* user_emb = (const float*)d_in[2];
    const float* item_emb = (const float*)d_in[3];
    const int*   adj_rows = (const int*)d_in[4];
    const int*   adj_cols = (const int*)d_in[5];
    const float* adj_vals = (const float*)d_in[6];
    float*       out      = (float*)d_out;

    const int       batch = in_sizes[0];
    const long long nU    = in_sizes[2] / EMB_D;
    const long long nI    = in_sizes[3] / EMB_D;
    const long long N     = nU + nI;
    const long long nnz   = in_sizes[4];
    const long long nElem = N * EMB_D;

    // Workspace: [cur | next | acc], 3 x N*64 floats (~115 MB total)
    float* bufA = (float*)d_ws;
    float* bufB = bufA + nElem;
    float* accb = bufB + nElem;

    const int       tpb   = 256;
    const long long nTot4 = nElem / 4;
    const long long nU4   = nU * EMB_D / 4;

    lgcn_init<<<(unsigned)((nTot4 + tpb - 1) / tpb), tpb, 0, stream>>>(
        (const float4*)user_emb, (const float4*)item_emb,
        (float4*)bufA, (float4*)accb, nU4, nTot4);

    float* cur = bufA;
    float* nxt = bufB;
    for (int layer = 0; layer < 3; ++layer) {
        hipMemsetAsync(nxt, 0, (size_t)nElem * sizeof(float), stream);

        long long nThreads = nnz * 16;
        lgcn_spmm_scatter<<<(unsigned)((nThreads + tpb - 1) / tpb), tpb, 0, stream>>>(
            adj_rows, adj_cols, adj_vals, cur, nxt, nnz);

        lgcn_acc_add<<<(unsigned)((nTot4 + tpb - 1) / tpb), tpb, 0, stream>>>(
            (float4*)accb, (const float4*)nxt, nTot4);

        float* tmp = cur; cur = nxt; nxt = tmp;
    }

    lgcn_dot_wmma<<<(unsigned)((batch + 15) / 16), 32, 0, stream>>>(
        users, items, accb, out, (int)nU, batch);
}